// BaselineVQVAE_81312320848238
// MI455X (gfx1250) — compile-verified
//
#include <hip/hip_runtime.h>
#include <hip/hip_bf16.h>

typedef float v2f __attribute__((ext_vector_type(2)));
typedef float v8f __attribute__((ext_vector_type(8)));

#define K_CODES   1024
#define DIM       64
#define HW        4096          // h*w
#define N_VEC     65536         // B*h*w
#define BETA_W    0.25f
#define DECAY_W   0.99f
#define EPS_V     1e-5f

// workspace layout in float units
#define WS_C2     0                       // [1024] codebook squared norms
#define WS_COUNTS 1024                    // [1024] segment counts
#define WS_SUMS   2048                    // [65536] segment sums (K x d)
#define WS_LOSS   (2048 + 65536)          // [1] SSE accumulator
#define WS_IDX    (WS_LOSS + 1)           // [65536] int32 indices

// ---------------------------------------------------------------------------
// Kernel 0: zero scratch, precompute ||c_k||^2
// ---------------------------------------------------------------------------
__global__ __launch_bounds__(256) void vq_init(const float* __restrict__ cb,
                                               float* __restrict__ ws) {
    int t = blockIdx.x * blockDim.x + threadIdx.x;
    if (t < N_VEC) ws[WS_SUMS + t] = 0.0f;
    if (t < K_CODES) {
        ws[WS_COUNTS + t] = 0.0f;
        const float* row = cb + t * DIM;
        float s = 0.0f;
        #pragma unroll 8
        for (int j = 0; j < DIM; ++j) { float v = row[j]; s += v * v; }
        ws[WS_C2 + t] = s;
    }
    if (t == 0) ws[WS_LOSS] = 0.0f;
}

// ---------------------------------------------------------------------------
// Kernel 1: nearest-codebook argmin via V_WMMA_F32_16X16X4_F32
//   score(m,k) = ||c_k||^2 - 2 * dot(z_m, c_k)   (||z||^2 drops out of argmin)
//   block = 8 waves; wave handles 16 rows; loop over 64 chunks of 16 codes.
// ---------------------------------------------------------------------------
__global__ __launch_bounds__(256) void vq_argmin(const float* __restrict__ z,
                                                 const float* __restrict__ cb,
                                                 const float* __restrict__ ws,
                                                 int* __restrict__ idx_ws,
                                                 float* __restrict__ idx_out) {
    __shared__ float lds_cb[16 * DIM];   // 16 codebook rows, row-major
    __shared__ float lds_c2[16];

    const int tid  = threadIdx.x;
    const int lane = tid & 31;
    const int wave = tid >> 5;
    const int half = lane >> 4;          // 0: K={0,1}, 1: K={2,3} within 4-chunk
    const int l16  = lane & 15;

    // global row owned by this lane in the A-matrix (M = lane&15)
    const int m = blockIdx.x * 128 + wave * 16 + l16;
    const float* zrow = z + (m >> 12) * (DIM * HW) + (m & (HW - 1));

    // Preload A: 16 steps x float2 (dims s*4 + half*2 + {0,1}), reused all chunks
    v2f a[16];
    #pragma unroll
    for (int s = 0; s < 16; ++s) {
        const int dim0 = s * 4 + half * 2;
        a[s].x = zrow[dim0 * HW];
        a[s].y = zrow[(dim0 + 1) * HW];
    }

    float best[8];
    int   bidx[8];
    #pragma unroll
    for (int r = 0; r < 8; ++r) { best[r] = 3.4e38f; bidx[r] = 0; }

    for (int chunk = 0; chunk < K_CODES / 16; ++chunk) {
        const int kbase = chunk * 16;
        __syncthreads();
        // 16 rows x 64 dims = 1024 contiguous floats -> 256 threads x float4
        ((float4*)lds_cb)[tid] = ((const float4*)(cb + kbase * DIM))[tid];
        if (tid < 16) lds_c2[tid] = ws[WS_C2 + kbase + tid];
        __syncthreads();

        v8f acc = {};
        const float* bbase = lds_cb + l16 * DIM + half * 2;  // B: N = lane&15
        #pragma unroll
        for (int s = 0; s < 16; ++s) {
            v2f bvec = *(const v2f*)(bbase + s * 4);
            acc = __builtin_amdgcn_wmma_f32_16x16x4_f32(
                false, a[s], false, bvec, (short)0, acc, false, false);
        }

        const float c2v  = lds_c2[l16];
        const int   kcol = kbase + l16;
        #pragma unroll
        for (int r = 0; r < 8; ++r) {
            float score = c2v - 2.0f * acc[r];
            if (score < best[r]) { best[r] = score; bidx[r] = kcol; }
        }
    }

    // min-reduce across the 16 columns held in each 16-lane half
    #pragma unroll
    for (int r = 0; r < 8; ++r) {
        #pragma unroll
        for (int off = 8; off > 0; off >>= 1) {
            float ov = __shfl_xor(best[r], off, 16);
            int   oi = __shfl_xor(bidx[r], off, 16);
            if (ov < best[r] || (ov == best[r] && oi < bidx[r])) {
                best[r] = ov; bidx[r] = oi;
            }
        }
    }

    if (l16 == 0) {
        // lanes 0/16 hold rows M = 0..7 / 8..15 of this wave's strip
        const int mt0 = blockIdx.x * 128 + wave * 16 + half * 8;
        #pragma unroll
        for (int r = 0; r < 8; ++r) {
            idx_ws[mt0 + r]  = bidx[r];
            idx_out[mt0 + r] = (float)bidx[r];
        }
    }
}

// ---------------------------------------------------------------------------
// Kernel 2: gather e_k (== e_k_ste numerically), SSE for L_commit,
//           fp32-atomic segment sums + counts.  One thread per (n, dim).
// ---------------------------------------------------------------------------
__global__ __launch_bounds__(256) void vq_scatter(const float* __restrict__ z,
                                                  const float* __restrict__ cb,
                                                  const int* __restrict__ idx_ws,
                                                  float* __restrict__ ws,
                                                  float* __restrict__ out_ek) {
    const int t   = blockIdx.x * blockDim.x + threadIdx.x;
    const int n   = t >> 6;
    const int dim = t & 63;
    const int k   = idx_ws[n];

    const float e = cb[k * DIM + dim];
    const int zoff = (n >> 12) * (DIM * HW) + dim * HW + (n & (HW - 1));
    const float zv = z[zoff];
    out_ek[zoff] = e;                           // straight-through == e_k

    float diff = zv - e;
    float sq = diff * diff;
    #pragma unroll
    for (int off = 16; off > 0; off >>= 1) sq += __shfl_xor(sq, off, 32);
    if ((threadIdx.x & 31) == 0) atomicAdd(&ws[WS_LOSS], sq);

    atomicAdd(&ws[WS_SUMS + k * DIM + dim], zv);
    if (dim == 0) atomicAdd(&ws[WS_COUNTS + k], 1.0f);
}

// ---------------------------------------------------------------------------
// Kernel 3: EMA update + Laplace smoothing + codebook normalize + loss write.
//   Single block of 1024 threads (one per code).
// ---------------------------------------------------------------------------
__global__ __launch_bounds__(1024) void vq_ema(const float* __restrict__ ema_count,
                                               const float* __restrict__ ema_sum,
                                               const float* __restrict__ ws,
                                               float* __restrict__ out_L,
                                               float* __restrict__ out_codebook,
                                               float* __restrict__ out_count,
                                               float* __restrict__ out_sum) {
    __shared__ float red[1024];
    const int k = threadIdx.x;

    float newc = ema_count[k] * DECAY_W + (1.0f - DECAY_W) * ws[WS_COUNTS + k];
    out_count[k] = newc;
    red[k] = newc;
    __syncthreads();
    for (int off = 512; off > 0; off >>= 1) {
        if (k < off) red[k] += red[k + off];
        __syncthreads();
    }
    const float n_total = red[0];
    const float count_s = (newc + EPS_V) / (n_total + (float)K_CODES * EPS_V) * n_total;
    const float inv = 1.0f / count_s;

    #pragma unroll 4
    for (int dim = 0; dim < DIM; ++dim) {
        float s = ema_sum[k * DIM + dim] * DECAY_W +
                  (1.0f - DECAY_W) * ws[WS_SUMS + k * DIM + dim];
        out_sum[k * DIM + dim]      = s;
        out_codebook[k * DIM + dim] = s * inv;
    }
    if (k == 0)
        out_L[0] = ws[WS_LOSS] * (1.0f + BETA_W) / (float)(N_VEC * DIM);
}

// ---------------------------------------------------------------------------
extern "C" void kernel_launch(void* const* d_in, const int* in_sizes, int n_in,
                              void* d_out, int out_size, void* d_ws, size_t ws_size,
                              hipStream_t stream) {
    const float* z         = (const float*)d_in[0];
    const float* cb        = (const float*)d_in[1];
    const float* ema_count = (const float*)d_in[2];
    const float* ema_sum   = (const float*)d_in[3];

    float* ws     = (float*)d_ws;
    int*   idx_ws = (int*)(ws + WS_IDX);

    float* out      = (float*)d_out;
    float* out_ek   = out;                         // [B,d,h,w] = 4194304
    float* out_idx  = out_ek + 4194304;            // [B,h,w]   = 65536
    float* out_L    = out_idx + 65536;             // [1]
    float* out_cbk  = out_L + 1;                   // [K,d]     = 65536
    float* out_cnt  = out_cbk + 65536;             // [K]       = 1024
    float* out_sum  = out_cnt + 1024;              // [K,d]     = 65536

    vq_init   <<<(N_VEC + 255) / 256, 256, 0, stream>>>(cb, ws);
    vq_argmin <<<N_VEC / 128,        256, 0, stream>>>(z, cb, ws, idx_ws, out_idx);
    vq_scatter<<<(N_VEC * DIM) / 256, 256, 0, stream>>>(z, cb, idx_ws, ws, out_ek);
    vq_ema    <<<1, 1024, 0, stream>>>(ema_count, ema_sum, ws, out_L,
                                       out_cbk, out_cnt, out_sum);
}